// Atte_module_77721728188895
// MI455X (gfx1250) — compile-verified
//
#include <hip/hip_runtime.h>

typedef __attribute__((ext_vector_type(2))) float v2f;
typedef __attribute__((ext_vector_type(8))) float v8f;

#define B_ 4
#define Q_ 256
#define K_ 256
#define H_ 768
#define D_ 768
#define NUMLEN 100

// ---------------------------------------------------------------------------
// proj_kernel: for each row x of X[nrows,H], compute
//   out1[row]     = x . W1  (+ b1 if addb)
//   out2[row*5+c] = x . W2[c] (+ b2[c] if addb)
// One block (256 threads) per row, LDS tree reduction.
// ---------------------------------------------------------------------------
__global__ void __launch_bounds__(256)
proj_kernel(const float* __restrict__ X, int nrows,
            const float* __restrict__ W1, const float* __restrict__ b1,
            const float* __restrict__ W2, const float* __restrict__ b2,
            int addb,
            float* __restrict__ out1, float* __restrict__ out2) {
    int row = blockIdx.x;
    if (row >= nrows) return;
    int tid = threadIdx.x;
    const float* x = X + (size_t)row * H_;

    float acc[6] = {0.f, 0.f, 0.f, 0.f, 0.f, 0.f};
    for (int h = tid; h < H_; h += 256) {
        float xv = x[h];
        acc[0] += xv * W1[h];
#pragma unroll
        for (int c = 0; c < 5; ++c) acc[1 + c] += xv * W2[c * H_ + h];
    }

    __shared__ float red[6][256];
#pragma unroll
    for (int j = 0; j < 6; ++j) red[j][tid] = acc[j];
    __syncthreads();
    for (int off = 128; off > 0; off >>= 1) {
        if (tid < off) {
#pragma unroll
            for (int j = 0; j < 6; ++j) red[j][tid] += red[j][tid + off];
        }
        __syncthreads();
    }
    if (tid == 0) {
        out1[row] = red[0][0] + (addb ? b1[0] : 0.f);
#pragma unroll
        for (int c = 0; c < 5; ++c)
            out2[row * 5 + c] = red[1 + c][0] + (addb ? b2[c] : 0.f);
    }
}

// ---------------------------------------------------------------------------
// scores_kernel: out_scores[b,q,k,c] = softmax_c( qW2[b,q,c]+kW2[b,k,c]+eW2[n,c] )
// qW2 already contains b2.  grid = B*Q blocks, 256 threads (k).
// ---------------------------------------------------------------------------
__global__ void __launch_bounds__(256)
scores_kernel(const int* __restrict__ lm,
              const float* __restrict__ qW2, const float* __restrict__ kW2,
              const float* __restrict__ eW2, float* __restrict__ out_scores) {
    int bq = blockIdx.x;          // b*Q + q
    int b  = bq / Q_;
    int k  = threadIdx.x;
    int idx = bq * K_ + k;        // flat (b,q,k)
    int n = lm[idx];

    float v[5];
    float mx = -3.402823466e38f;
#pragma unroll
    for (int c = 0; c < 5; ++c) {
        v[c] = qW2[bq * 5 + c] + kW2[(b * K_ + k) * 5 + c] + eW2[n * 5 + c];
        mx = fmaxf(mx, v[c]);
    }
    float sum = 0.f;
#pragma unroll
    for (int c = 0; c < 5; ++c) {
        v[c] = __expf(v[c] - mx);
        sum += v[c];
    }
    float inv = 1.f / sum;
#pragma unroll
    for (int c = 0; c < 5; ++c)
        out_scores[(size_t)idx * 5 + c] = v[c] * inv;
}

// ---------------------------------------------------------------------------
// probs_kernel: probs[b,q,k] = softmax_k( (qW1[b,q]+kW1[b,k]+eW1[lm]) * mask )
// qW1 already contains b1 (added before the multiplicative mask, matching ref).
// grid = B*Q blocks, 256 threads (k).
// ---------------------------------------------------------------------------
__global__ void __launch_bounds__(256)
probs_kernel(const int* __restrict__ lm, const float* __restrict__ am,
             const float* __restrict__ qW1, const float* __restrict__ kW1,
             const float* __restrict__ eW1, float* __restrict__ probs) {
    int bq = blockIdx.x;
    int b  = bq / Q_;
    int k  = threadIdx.x;
    int idx = bq * K_ + k;

    float s = (qW1[bq] + kW1[b * K_ + k] + eW1[lm[idx]]) * am[idx];

    __shared__ float red[256];
    red[k] = s;
    __syncthreads();
    for (int off = 128; off > 0; off >>= 1) {
        if (k < off) red[k] = fmaxf(red[k], red[k + off]);
        __syncthreads();
    }
    float mx = red[0];
    __syncthreads();

    float e = __expf(s - mx);
    red[k] = e;
    __syncthreads();
    for (int off = 128; off > 0; off >>= 1) {
        if (k < off) red[k] += red[k + off];
        __syncthreads();
    }
    probs[idx] = e * (1.f / red[0]);
}

// ---------------------------------------------------------------------------
// content_gemm_kernel: C[b] = P[b](256x256) * M[b](256x768) in f32 via
// V_WMMA_F32_16X16X4_F32, with the m panel staged into LDS by
// GLOBAL_LOAD_ASYNC_TO_LDS_B32 (ASYNCcnt double-buffer pipeline).
//
// Block = 256 threads (8 waves), computes a 16(q) x 128(d) output tile:
//  - P tile (16 x 256 probs) loaded cooperatively to LDS once.
//  - Each wave owns a 16-wide d slab; it async-copies its own slab of m,
//    TRANSPOSED on the fly ([k][d] global -> [d][k] LDS, per-lane scatter
//    addresses), in chunks of 32 k-rows, double-buffered on ASYNCcnt.
//    Async loads retire in order, so after issuing the next chunk's 16 ops,
//    s_wait_asynccnt 16 guarantees the current chunk is resident.
//  - B fragment per WMMA step is then a single aligned ds_load_b64
//    (two adjacent k entries of one d column).
//  A frag (16x4 f32, 2 VGPR): lanes 0-15 -> M=lane, K={kk,kk+1};
//                             lanes 16-31 -> M=lane-16, K={kk+2,kk+3}.
//  B frag (4x16 f32, 2 VGPR): lanes 0-15 -> N=lane, rows {kk,kk+1};
//                             lanes 16-31 -> N=lane-16, rows {kk+2,kk+3}.
//  C/D (16x16 f32, 8 VGPR): VGPR r, lanes 0-15 -> row r; lanes 16-31 -> r+8.
// grid = (B*Q/16, D/128)
// ---------------------------------------------------------------------------
#define KCH  32     // k rows per staged chunk
#define TSTR 34     // padded per-column k stride in dwords (bank spread, even)

__global__ void __launch_bounds__(256)
content_gemm_kernel(const float* __restrict__ P,   // [B,Q,K] probs
                    const float* __restrict__ M,   // [B,K,D]
                    float* __restrict__ C) {       // [B,Q,D]
    __shared__ float Ptile[16][K_ + 4];
    __shared__ float Mt[2][128][TSTR];   // [buf][block-local d col][k in chunk]

    int bq16 = blockIdx.x;            // b*(Q/16) + qtile
    int b    = bq16 / (Q_ / 16);
    int q0   = (bq16 % (Q_ / 16)) * 16;
    int d0   = blockIdx.y * 128;

    int tid  = threadIdx.x;
    int wave = tid >> 5;              // 0..7 -> d sub-tile
    int lane = tid & 31;
    int half = lane >> 4;             // 0 or 1
    int l16  = lane & 15;

    const float* Pb = P + ((size_t)b * Q_ + q0) * K_;
    const float* Mb = M + (size_t)b * K_ * D_;
    int dcol = d0 + wave * 16;        // global first d column of this wave's slab
    int col  = wave * 16 + l16;       // block-local column index into Mt

    unsigned mt_base = (unsigned)(unsigned long long)(uintptr_t)&Mt[0][0][0];

    // Async-stage one 32-row k-chunk of this wave's 16 m columns, transposing
    // into LDS.  16 b32 async ops per lane (lane covers column l16, rows
    // half*16 .. half*16+15).
    auto issue_chunk = [&](int ch, int bufi) {
        int r0 = half * 16;
        const float* g0 = Mb + (size_t)(ch * KCH + r0) * D_ + dcol + l16;
        unsigned lds0 = mt_base +
            ((unsigned)(bufi * 128 + col) * TSTR + (unsigned)r0) * 4u;
#pragma unroll
        for (int j = 0; j < 16; ++j) {
            unsigned long long g = (unsigned long long)(uintptr_t)(g0 + (size_t)j * D_);
            unsigned lds = lds0 + (unsigned)j * 4u;
            asm volatile("global_load_async_to_lds_b32 %0, %1, off"
                         :: "v"(lds), "v"(g) : "memory");
        }
    };

    // Kick off chunk 0 before touching Ptile so the DMA overlaps the staging.
    issue_chunk(0, 0);

    for (int i = tid; i < 16 * K_; i += 256) {
        int r = i >> 8;
        int c = i & (K_ - 1);
        Ptile[r][c] = Pb[r * K_ + c];
    }
    __syncthreads();

    v8f acc = {0.f, 0.f, 0.f, 0.f, 0.f, 0.f, 0.f, 0.f};
    int buf = 0;
    for (int ch = 0; ch < K_ / KCH; ++ch) {
        if (ch + 1 < K_ / KCH) {
            issue_chunk(ch + 1, buf ^ 1);
            // 16 ops of the next chunk outstanding; in-order retire means
            // ASYNCcnt<=16 <=> current chunk fully resident in LDS.
            asm volatile("s_wait_asynccnt 16" ::: "memory");
        } else {
            asm volatile("s_wait_asynccnt 0" ::: "memory");
        }
        if (ch + 2 < K_ / KCH)   // pull chunk-after-next toward L2
            __builtin_prefetch(Mb + (size_t)((ch + 2) * KCH + half * 16) * D_ + dcol + l16,
                               0, 1);

#pragma unroll
        for (int kk = 0; kk < KCH; kk += 4) {
            v2f a;
            a.x = Ptile[l16][ch * KCH + kk + 2 * half];
            a.y = Ptile[l16][ch * KCH + kk + 2 * half + 1];

            v2f bf;   // one aligned ds_load_b64: adjacent k entries of column
            bf.x = Mt[buf][col][kk + 2 * half];
            bf.y = Mt[buf][col][kk + 2 * half + 1];

            acc = __builtin_amdgcn_wmma_f32_16x16x4_f32(
                /*neg_a=*/false, a, /*neg_b=*/false, bf,
                /*c_mod=*/(short)0, acc, /*reuse_a=*/false, /*reuse_b=*/false);
        }
        buf ^= 1;
    }

    float* Cb = C + ((size_t)b * Q_ + q0) * D_;
#pragma unroll
    for (int r = 0; r < 8; ++r) {
        int row = r + half * 8;
        Cb[(size_t)row * D_ + dcol + l16] = acc[r];
    }
}

// ---------------------------------------------------------------------------
extern "C" void kernel_launch(void* const* d_in, const int* in_sizes, int n_in,
                              void* d_out, int out_size, void* d_ws, size_t ws_size,
                              hipStream_t stream) {
    (void)in_sizes; (void)n_in; (void)out_size; (void)ws_size;

    const float* q   = (const float*)d_in[0];   // [B,Q,H]
    const float* k   = (const float*)d_in[1];   // [B,K,H]
    const float* m   = (const float*)d_in[2];   // [B,K,D]
    const float* am  = (const float*)d_in[3];   // [B,Q,K]
    const float* emb = (const float*)d_in[4];   // [NUMLEN,H]
    const float* W1  = (const float*)d_in[5];   // [1,H]
    const float* b1  = (const float*)d_in[6];   // [1]
    const float* W2  = (const float*)d_in[7];   // [5,H]
    const float* b2  = (const float*)d_in[8];   // [5]
    const int*   lm  = (const int*)d_in[9];     // [B,Q,K]

    float* out_content = (float*)d_out;                       // [B,Q,D]
    float* out_scores  = out_content + (size_t)B_ * Q_ * D_;  // [B,Q,K,5]

    float* ws    = (float*)d_ws;
    float* qW1   = ws;               // 1024
    float* qW2   = qW1 + 1024;       // 5120
    float* kW1   = qW2 + 5120;       // 1024
    float* kW2   = kW1 + 1024;       // 5120
    float* eW1   = kW2 + 5120;       // 128 (100 used)
    float* eW2   = eW1 + 128;        // 512 (500 used)
    float* probs = eW2 + 512;        // B*Q*K = 262144

    proj_kernel<<<B_ * Q_, 256, 0, stream>>>(q,   B_ * Q_, W1, b1, W2, b2, 1, qW1, qW2);
    proj_kernel<<<B_ * K_, 256, 0, stream>>>(k,   B_ * K_, W1, b1, W2, b2, 0, kW1, kW2);
    proj_kernel<<<NUMLEN,  256, 0, stream>>>(emb, NUMLEN,  W1, b1, W2, b2, 0, eW1, eW2);

    scores_kernel<<<B_ * Q_, 256, 0, stream>>>(lm, qW2, kW2, eW2, out_scores);
    probs_kernel <<<B_ * Q_, 256, 0, stream>>>(lm, am, qW1, kW1, eW1, probs);

    dim3 g(B_ * (Q_ / 16), D_ / 128);
    content_gemm_kernel<<<g, 256, 0, stream>>>(probs, m, out_content);
}